// GAT_70540542869744
// MI455X (gfx1250) — compile-verified
//
#include <hip/hip_runtime.h>
#include <hip/hip_bf16.h>
#include <math.h>

#define NN   50000
#define EE   800000
#define IND  128
#define HID  32
#define NH   4
#define NC   40
#define FDIM 128   // H*HID

typedef float v8f __attribute__((ext_vector_type(8)));
typedef float v2f __attribute__((ext_vector_type(2)));

// ---------------- monotone float <-> uint key for atomic max ----------------
__device__ __forceinline__ unsigned fkey(float f) {
    unsigned b = __float_as_uint(f);
    return (b & 0x80000000u) ? ~b : (b | 0x80000000u);
}
__device__ __forceinline__ float funkey(unsigned k) {
    unsigned b = (k & 0x80000000u) ? (k & 0x7FFFFFFFu) : ~k;
    return __uint_as_float(b);
}

// ---------------- GEMM: Y[N,128] = X[N,128] @ W[128,128] (WMMA f32) ---------
// block = 256 threads (8 waves); block handles a 16-row tile, wave w -> cols 16w..16w+15
__global__ void gat_gemm128(const float* __restrict__ X,
                            const float* __restrict__ W,
                            float* __restrict__ Y) {
    __shared__ float As[16 * FDIM];   // 8 KB
    const int m0 = blockIdx.x * 16;
    const int t  = threadIdx.x;

    // stage A tile (2048 floats) via float4
    for (int i = t; i < 16 * FDIM / 4; i += 256)
        ((float4*)As)[i] = ((const float4*)(X + (size_t)m0 * FDIM))[i];
    __syncthreads();

    const int wave  = t >> 5;
    const int lane  = t & 31;
    const int n0    = wave * 16;
    const int mrow  = lane & 15;
    const int kgrp  = lane >> 4;      // 0 or 1

    v8f acc = {};
    for (int kk = 0; kk < FDIM; kk += 4) {
        v2f a, b;
        // A 16x4: lane m=mrow, VGPR j -> K = kk + 2*kgrp + j
        a.x = As[mrow * FDIM + kk + 2 * kgrp + 0];
        a.y = As[mrow * FDIM + kk + 2 * kgrp + 1];
        // B 4x16: lane col = mrow, VGPR j -> K = kk + 2*kgrp + j
        b.x = W[(kk + 2 * kgrp + 0) * FDIM + n0 + mrow];
        b.y = W[(kk + 2 * kgrp + 1) * FDIM + n0 + mrow];
        acc = __builtin_amdgcn_wmma_f32_16x16x4_f32(false, a, false, b,
                                                    (short)0, acc, false, false);
    }
    // C 16x16 f32: VGPR r -> M = 8*(lane>>4)+r, N = lane&15
    const int cn    = lane & 15;
    const int mbase = (lane >> 4) * 8;
    for (int r = 0; r < 8; ++r)
        Y[(size_t)(m0 + mbase + r) * FDIM + n0 + cn] = acc[r];
}

// ---------------- output GEMM: logits[N,40] = Hm[N,128] @ Wout[128,40] + b --
// block = 96 threads (3 waves) covering cols 0..47, masked at 40
__global__ void gat_gemm_out(const float* __restrict__ Hm,
                             const float* __restrict__ Wout,
                             const float* __restrict__ bout,
                             float* __restrict__ logits) {
    __shared__ float As[16 * FDIM];
    const int m0 = blockIdx.x * 16;
    const int t  = threadIdx.x;
    for (int i = t; i < 16 * FDIM / 4; i += 96)
        ((float4*)As)[i] = ((const float4*)(Hm + (size_t)m0 * FDIM))[i];
    __syncthreads();

    const int wave = t >> 5;
    const int lane = t & 31;
    const int n0   = wave * 16;
    const int mrow = lane & 15;
    const int kgrp = lane >> 4;
    const int col  = n0 + mrow;

    v8f acc = {};
    for (int kk = 0; kk < FDIM; kk += 4) {
        v2f a, b;
        a.x = As[mrow * FDIM + kk + 2 * kgrp + 0];
        a.y = As[mrow * FDIM + kk + 2 * kgrp + 1];
        b.x = (col < NC) ? Wout[(kk + 2 * kgrp + 0) * NC + col] : 0.0f;
        b.y = (col < NC) ? Wout[(kk + 2 * kgrp + 1) * NC + col] : 0.0f;
        acc = __builtin_amdgcn_wmma_f32_16x16x4_f32(false, a, false, b,
                                                    (short)0, acc, false, false);
    }
    const int cn    = lane & 15;
    const int mbase = (lane >> 4) * 8;
    const int ocol  = n0 + cn;
    if (ocol < NC) {
        const float bb = bout[ocol];
        for (int r = 0; r < 8; ++r)
            logits[(size_t)(m0 + mbase + r) * NC + ocol] = acc[r] + bb;
    }
}

// ---------------- el/er: per (node, head) dot over 32 dims ------------------
__global__ void gat_attn_coeff(const float* __restrict__ feat,
                               const float* __restrict__ al,
                               const float* __restrict__ ar,
                               float* __restrict__ el,
                               float* __restrict__ er) {
    int i = blockIdx.x * blockDim.x + threadIdx.x;   // over N*H
    if (i >= NN * NH) return;
    int n = i >> 2, h = i & 3;
    const float* f = feat + (size_t)n * FDIM + h * HID;
    float sl = 0.f, sr = 0.f;
    for (int d = 0; d < HID; ++d) {
        sl += f[d] * al[h * HID + d];
        sr += f[d] * ar[h * HID + d];
    }
    el[i] = sl;
    er[i] = sr;
}

// ---------------- edge logits + segment max ---------------------------------
__global__ void gat_edge_logits(const float* __restrict__ el,
                                const float* __restrict__ er,
                                const int* __restrict__ src,
                                const int* __restrict__ dst,
                                float* __restrict__ ebuf,
                                unsigned* __restrict__ maxkey) {
    int i = blockIdx.x * blockDim.x + threadIdx.x;   // over E*H
    if (i >= EE * NH) return;
    int e = i >> 2, h = i & 3;
    int s = src[e], d = dst[e];
    float v = el[s * NH + h] + er[d * NH + h];
    v = (v > 0.f) ? v : 0.2f * v;                    // leaky relu, slope 0.2
    ebuf[i] = v;
    atomicMax(&maxkey[d * NH + h], fkey(v));
}

// ---------------- exp(e - max) + segment sum --------------------------------
__global__ void gat_edge_exp(const int* __restrict__ dst,
                             float* __restrict__ ebuf,
                             const unsigned* __restrict__ maxkey,
                             float* __restrict__ denom) {
    int i = blockIdx.x * blockDim.x + threadIdx.x;   // over E*H
    if (i >= EE * NH) return;
    int e = i >> 2, h = i & 3;
    int d = dst[e];
    float m  = funkey(maxkey[d * NH + h]);
    float ee = expf(ebuf[i] - m);
    ebuf[i] = ee;                                    // reuse buffer for ee
    atomicAdd(&denom[d * NH + h], ee);
}

// ---------------- scatter messages: agg[dst] += feat[src] * a ---------------
__global__ void gat_scatter(const int* __restrict__ src,
                            const int* __restrict__ dst,
                            const float* __restrict__ ebuf,
                            const float* __restrict__ denom,
                            const float* __restrict__ feat,
                            float* __restrict__ agg) {
    int i = blockIdx.x * blockDim.x + threadIdx.x;   // over E*128
    if (i >= EE * FDIM) return;
    int e = i >> 7;
    int c = i & (FDIM - 1);
    int h = c >> 5;
    int s = src[e], d = dst[e];
    float a = ebuf[e * NH + h] / denom[d * NH + h];
    atomicAdd(&agg[(size_t)d * FDIM + c], feat[(size_t)s * FDIM + c] * a);
}

// ---------------- elu (+ optional residual) ---------------------------------
__global__ void gat_elu(const float* agg, const float* resid, float* out, int use_res) {
    int i = blockIdx.x * blockDim.x + threadIdx.x;   // over N*128
    if (i >= NN * FDIM) return;
    float v = agg[i] + (use_res ? resid[i] : 0.f);
    out[i] = (v > 0.f) ? v : (expf(v) - 1.0f);
}

// ---------------- mixup: hout = lam*hout + (1-lam)*h2 (in place) ------------
__global__ void gat_mix(const float* __restrict__ h2,
                        const float* __restrict__ lamb,
                        float* hout) {
    int i = blockIdx.x * blockDim.x + threadIdx.x;
    if (i >= NN * FDIM) return;
    float lam = lamb[0];
    hout[i] = lam * hout[i] + (1.0f - lam) * h2[i];
}

// ---------------- one GAT conv layer ----------------------------------------
static void run_conv(const float* X, const float* W, const float* al, const float* ar,
                     const int* src, const int* dst, const float* resid, float* out,
                     float* feat, float* agg, float* ebuf, float* el, float* er,
                     unsigned* maxkey, float* denom, hipStream_t stream) {
    hipMemsetAsync(agg,    0, (size_t)NN * FDIM * sizeof(float), stream);
    hipMemsetAsync(maxkey, 0, (size_t)NN * NH * sizeof(unsigned), stream);  // key 0 == -max
    hipMemsetAsync(denom,  0, (size_t)NN * NH * sizeof(float), stream);

    gat_gemm128<<<NN / 16, 256, 0, stream>>>(X, W, feat);
    gat_attn_coeff<<<(NN * NH + 255) / 256, 256, 0, stream>>>(feat, al, ar, el, er);
    gat_edge_logits<<<(EE * NH + 255) / 256, 256, 0, stream>>>(el, er, src, dst, ebuf, maxkey);
    gat_edge_exp<<<(EE * NH + 255) / 256, 256, 0, stream>>>(dst, ebuf, maxkey, denom);
    gat_scatter<<<(EE * FDIM + 255) / 256, 256, 0, stream>>>(src, dst, ebuf, denom, feat, agg);
    gat_elu<<<(NN * FDIM + 255) / 256, 256, 0, stream>>>(agg, resid, out, resid != nullptr);
}

extern "C" void kernel_launch(void* const* d_in, const int* in_sizes, int n_in,
                              void* d_out, int out_size, void* d_ws, size_t ws_size,
                              hipStream_t stream) {
    (void)in_sizes; (void)n_in; (void)out_size; (void)ws_size;

    const float* inputs = (const float*)d_in[0];
    const float* target = (const float*)d_in[1];
    const float* lamb   = (const float*)d_in[2];
    const float* W0     = (const float*)d_in[3];
    const float* al0    = (const float*)d_in[4];
    const float* ar0    = (const float*)d_in[5];
    const float* W1     = (const float*)d_in[6];
    const float* al1    = (const float*)d_in[7];
    const float* ar1    = (const float*)d_in[8];
    const float* Wout   = (const float*)d_in[9];
    const float* bout   = (const float*)d_in[10];
    const int*   src    = (const int*)d_in[11];
    const int*   dst    = (const int*)d_in[12];

    // workspace carve (floats)
    float* feat   = (float*)d_ws;                      // N*128
    float* agg    = feat + (size_t)NN * FDIM;          // N*128
    float* hbuf   = agg  + (size_t)NN * FDIM;          // N*128
    float* ebuf   = hbuf + (size_t)NN * FDIM;          // E*4
    float* el     = ebuf + (size_t)EE * NH;            // N*4
    float* er     = el   + (size_t)NN * NH;            // N*4
    unsigned* mk  = (unsigned*)(er + (size_t)NN * NH); // N*4
    float* denom  = (float*)(mk + (size_t)NN * NH);    // N*4

    float* hregion = (float*)d_out;                    // [N,128] h output
    float* logits  = hregion + (size_t)NN * FDIM;      // [N,40]

    // path 1 (inputs): layer0 -> hbuf, layer1 (+residual) -> d_out h-region
    run_conv(inputs, W0, al0, ar0, src, dst, nullptr, hbuf,
             feat, agg, ebuf, el, er, mk, denom, stream);
    run_conv(hbuf, W1, al1, ar1, src, dst, hbuf, hregion,
             feat, agg, ebuf, el, er, mk, denom, stream);

    // path 2 (target): layer0 -> hbuf, layer1 (+residual) -> hbuf (in place)
    run_conv(target, W0, al0, ar0, src, dst, nullptr, hbuf,
             feat, agg, ebuf, el, er, mk, denom, stream);
    run_conv(hbuf, W1, al1, ar1, src, dst, hbuf, hbuf,
             feat, agg, ebuf, el, er, mk, denom, stream);

    // mixup (in place on d_out h-region), then output projection
    gat_mix<<<(NN * FDIM + 255) / 256, 256, 0, stream>>>(hbuf, lamb, hregion);
    gat_gemm_out<<<NN / 16, 96, 0, stream>>>(hregion, Wout, bout, logits);
}